// NodeSelection_12575664243035
// MI455X (gfx1250) — compile-verified
//
#include <hip/hip_runtime.h>
#include <hip/hip_bf16.h>
#include <math.h>

// Problem constants (from reference setup_inputs)
#define NN 128
#define MM 4096
#define DD 256
#define FF 128
#define KK 32

typedef __attribute__((ext_vector_type(2))) float v2f;
typedef __attribute__((ext_vector_type(8))) float v8f;

// ---------------------------------------------------------------------------
// Kernel 1: fused cosine-score pass.
//   scores[n][m] = dot(emb[n,m,:], emb[n,tgt[n],:]) * rsqrt(||emb[n,m,:]||^2)
// (division by ||tgt|| is a positive per-n constant -> does not change top-k)
// One wave computes a 16-row tile via V_WMMA_F32_16X16X4_F32 over D=256 in
// 64 K-steps of 4. Norms ride along on the same A-matrix loads.
// grid = (N, 32), block = 256 (8 waves, each wave -> one 16-row tile)
// ---------------------------------------------------------------------------
__global__ __launch_bounds__(256)
void ns_score_kernel(const float* __restrict__ emb,
                     const int*   __restrict__ tgt,
                     float*       __restrict__ scores) {
    const int n    = blockIdx.x;
    const int lane = threadIdx.x & 31;
    const int wave = threadIdx.x >> 5;

    __shared__ float tgt_lds[DD];

    const float* embn = emb + (size_t)n * MM * DD;
    const int tnode = tgt[n];

    // Stage target embedding (256 floats) in LDS: one float per thread.
    tgt_lds[threadIdx.x] = embn[(size_t)tnode * DD + threadIdx.x];
    __syncthreads();

    const int tile  = blockIdx.y * 8 + wave;     // 0..255
    const int mbase = tile * 16;
    const int mrow  = mbase + (lane & 15);       // A-matrix M for this lane
    const int khalf = (lane >> 4) * 2;           // lanes 0-15 -> K+0,1 ; 16-31 -> K+2,3

    const float* __restrict__ rowp = embn + (size_t)mrow * DD;

    v8f  c   = {};        // 16x16 f32 accumulator (all columns identical)
    float nsq = 0.0f;     // per-lane-half partial ||row||^2

    #pragma unroll 4
    for (int k = 0; k < DD; k += 4) {
        v2f a = *(const v2f*)(rowp + k + khalf);           // A[M=lane%16][k+khalf ..]
        v2f b = *(const v2f*)(&tgt_lds[k + khalf]);        // B col-replicated tgt chunk
        nsq += a.x * a.x + a.y * a.y;
        // D = A(16x4) x B(4x16) + C   -> v_wmma_f32_16x16x4_f32
        c = __builtin_amdgcn_wmma_f32_16x16x4_f32(
                /*neg_a=*/false, a, /*neg_b=*/false, b,
                /*c_mod=*/(short)0, c, /*reuse_a=*/false, /*reuse_b=*/false);
    }

    // Combine the two K-halves of each row's norm: lanes L and L+16 share row L.
    nsq += __shfl_xor(nsq, 16, 32);

    // Unpack accumulator: lanes 0-15 hold sims for rows mbase+0..7 (c[0..7]),
    // lanes 16-31 hold sims for rows mbase+8..15 (column replication).
    float carr[8];
    #pragma unroll
    for (int r = 0; r < 8; ++r) carr[r] = c[r];

    // Writers: lanes 0..7 -> rows 0..7 ; lanes 16..23 -> rows 8..15.
    const int j   = lane & 7;
    const int row = (lane < 16) ? j : (8 + j);
    // nsq for row `row` lives at lane `row` (and row+16); fetch it.
    const float nsq_row = __shfl(nsq, (lane < 16) ? lane : (lane - 8), 32);
    const float val = carr[j] * rsqrtf(nsq_row);

    const bool writer = (lane < 8) || (lane >= 16 && lane < 24);
    if (writer) {
        scores[(size_t)n * MM + mbase + row] = val;
    }
}

// ---------------------------------------------------------------------------
// Kernel 2: top-32 per n (JAX semantics: descending value, ties -> lower index).
// One block of 256 threads per n; scores staged in LDS; 32 argmax rounds.
// ---------------------------------------------------------------------------
__global__ __launch_bounds__(256)
void ns_topk_kernel(const float* __restrict__ scores,
                    int*         __restrict__ topk) {
    const int n   = blockIdx.x;
    const int tid = threadIdx.x;

    __shared__ float sval[MM];
    __shared__ float redv[256];
    __shared__ int   redi[256];

    for (int i = tid; i < MM; i += 256)
        sval[i] = scores[(size_t)n * MM + i];
    __syncthreads();

    for (int k = 0; k < KK; ++k) {
        float bv = -INFINITY;
        int   bi = MM;                       // sentinel larger than any index
        for (int i = tid; i < MM; i += 256) {
            float v = sval[i];
            if (v > bv) { bv = v; bi = i; }  // strided i increases -> ties impossible per-thread
        }
        redv[tid] = bv;
        redi[tid] = bi;
        __syncthreads();
        for (int s = 128; s > 0; s >>= 1) {
            if (tid < s) {
                float ov = redv[tid + s];
                int   oi = redi[tid + s];
                if (ov > redv[tid] || (ov == redv[tid] && oi < redi[tid])) {
                    redv[tid] = ov;
                    redi[tid] = oi;
                }
            }
            __syncthreads();
        }
        if (tid == 0) {
            topk[n * KK + k] = redi[0];
            sval[redi[0]] = -INFINITY;       // remove winner
        }
        __syncthreads();
    }
}

// ---------------------------------------------------------------------------
// Kernel 3: gather selected rows into concatenated output:
//   out[0 .. N*K*F)           = x_sel   (N,32,128)
//   out[N*K*F .. +N*K*D)      = emb_sel (N,32,256)
// grid = N*K blocks, 128 threads each.
// ---------------------------------------------------------------------------
__global__ __launch_bounds__(128)
void ns_gather_kernel(const float* __restrict__ x,
                      const float* __restrict__ emb,
                      const int*   __restrict__ topk,
                      float*       __restrict__ out) {
    const int b = blockIdx.x;          // 0 .. N*K-1
    const int n = b >> 5;
    const int k = b & 31;
    const int m = topk[b];
    const int t = threadIdx.x;         // 0..127

    const float* __restrict__ xrow = x + ((size_t)n * MM + m) * FF;
    float* __restrict__ xout = out + ((size_t)n * KK + k) * FF;
    xout[t] = xrow[t];

    const float* __restrict__ erow = emb + ((size_t)n * MM + m) * DD;
    float* __restrict__ eout = out + (size_t)NN * KK * FF + ((size_t)n * KK + k) * DD;
    eout[t]       = erow[t];
    eout[t + 128] = erow[t + 128];
}

// ---------------------------------------------------------------------------
extern "C" void kernel_launch(void* const* d_in, const int* in_sizes, int n_in,
                              void* d_out, int out_size, void* d_ws, size_t ws_size,
                              hipStream_t stream) {
    const float* x   = (const float*)d_in[0];   // (N, M, F) f32
    const float* emb = (const float*)d_in[1];   // (N, M, D) f32
    const int*   tgt = (const int*)  d_in[2];   // (N,) int
    float*       out = (float*)d_out;

    // Workspace layout: scores (N*M f32 = 2 MB) | topk (N*K int = 16 KB)
    float* scores = (float*)d_ws;
    int*   topk   = (int*)((char*)d_ws + (size_t)NN * MM * sizeof(float));

    dim3 g1(NN, MM / 16 / 8);   // (128, 32): 8 wave-tiles of 16 rows per block
    ns_score_kernel<<<g1, 256, 0, stream>>>(emb, tgt, scores);

    ns_topk_kernel<<<NN, 256, 0, stream>>>(scores, topk);

    ns_gather_kernel<<<NN * KK, 128, 0, stream>>>(x, emb, topk, out);
}